// PhaseAwareBinaryAttention_50122268344525
// MI455X (gfx1250) — compile-verified
//
#include <hip/hip_runtime.h>

typedef __attribute__((ext_vector_type(2))) float v2f;
typedef __attribute__((ext_vector_type(8))) float v8f;

// ---------------------------------------------------------------------------
// Kernel 1: per-batch threshold via closed-form pair statistics.
// sum_{i,j} dist and sum_{i,j} dist^2 expressed in 12 point moments (fp64).
// One block per batch, reduction over S=4096 points only (vs S^2 pairs).
// ---------------------------------------------------------------------------
__global__ __launch_bounds__(256) void paba_stats_kernel(
    const float2* __restrict__ r, float* __restrict__ thr_out, int S) {
  const int b = blockIdx.x;
  const float2* p = r + (long long)b * S;

  double m[12];
#pragma unroll
  for (int k = 0; k < 12; ++k) m[k] = 0.0;

  for (int s = threadIdx.x; s < S; s += blockDim.x) {
    float2 v = p[s];
    double x = (double)v.x, y = (double)v.y;
    double x2 = x * x, y2 = y * y;
    m[0]  += x;       m[1]  += y;
    m[2]  += x2;      m[3]  += y2;
    m[4]  += x2 * x;  m[5]  += y2 * y;
    m[6]  += x2 * x2; m[7]  += y2 * y2;
    m[8]  += x * y;   m[9]  += x2 * y;
    m[10] += x * y2;  m[11] += x2 * y2;
  }

  __shared__ double red[256];
  double tot[12];
#pragma unroll
  for (int k = 0; k < 12; ++k) {
    red[threadIdx.x] = m[k];
    __syncthreads();
    for (int off = 128; off > 0; off >>= 1) {
      if ((int)threadIdx.x < off) red[threadIdx.x] += red[threadIdx.x + off];
      __syncthreads();
    }
    tot[k] = red[0];
    __syncthreads();
  }

  if (threadIdx.x == 0) {
    double Sx = tot[0], Sy = tot[1], Sxx = tot[2], Syy = tot[3];
    double Sx3 = tot[4], Sy3 = tot[5], Sx4 = tot[6], Sy4 = tot[7];
    double Sxy = tot[8], Sx2y = tot[9], Sxy2 = tot[10], Sx2y2 = tot[11];
    double Sd = (double)S;
    double N = Sd * Sd;
    // sum over all pairs of dist
    double sum1 = 2.0 * Sd * (Sxx + Syy) - 2.0 * (Sx * Sx + Sy * Sy);
    // sum over all pairs of dist^2
    double sdx4 = 2.0 * Sd * Sx4 - 8.0 * Sx3 * Sx + 6.0 * Sxx * Sxx;
    double sdy4 = 2.0 * Sd * Sy4 - 8.0 * Sy3 * Sy + 6.0 * Syy * Syy;
    double sdx2dy2 = 2.0 * Sd * Sx2y2 - 4.0 * Sy * Sx2y - 4.0 * Sx * Sxy2 +
                     2.0 * Sxx * Syy + 4.0 * Sxy * Sxy;
    double sum2 = sdx4 + 2.0 * sdx2dy2 + sdy4;
    double mean = sum1 / N;
    double var = (sum2 - sum1 * sum1 / N) / (N - 1.0);  // ddof=1
    if (var < 0.0) var = 0.0;
    thr_out[b] = (float)(mean + 1.25 * sqrt(var));
  }
}

// ---------------------------------------------------------------------------
// Tile epilogue: dx/dy tiles come straight out of the WMMA accumulators.
// DIAG selects whether the per-element causal compare is needed.
// ---------------------------------------------------------------------------
template <bool DIAG>
__device__ __forceinline__ void paba_epilogue(const v8f adx, const v8f ady,
                                              float thr, float* __restrict__ outT,
                                              int S, int csel, int mhi) {
#pragma unroll
  for (int rr = 0; rr < 8; ++rr) {
    const float dx = adx[rr];
    const float dy = ady[rr];
    const float dist = fmaf(dx, dx, dy * dy);        // exact per-ref dist_sq
    const float inv = __builtin_amdgcn_rsqf(dist);   // v_rsq_f32
    float bias = fmaf(dx * inv, 0.5f, 0.5f);         // 0.5*(1+cos(atan2(dy,dx)))
    const bool zp = (dx == 0.0f) && (dy == 0.0f);
    bias = zp ? 1.0f : bias;                         // cos(atan2(0,1)) = 1
    bool valid = (dist <= thr);                      // zp => dist==0 <= thr
    if (DIAG) valid = valid && ((rr + mhi) >= csel); // causal (i >= j)
    outT[(long long)(rr + mhi) * S + csel] = valid ? bias : 0.0f;
  }
}

// ---------------------------------------------------------------------------
// Kernel 2: one wave per 16x16 output tile.
//   dx[m][n] = x_i[m] - x_j[n] = [x_i,1] x [1,-x_j]: rank-2 outer difference
//   built by V_WMMA_F32_16X16X4_F32 (exact: products are *1.0, one fp add).
//   Second WMMA builds dy. No LDS/shuffle traffic at all.
//   Strictly-upper tiles: pure zero fill with b128 stores.
// ---------------------------------------------------------------------------
__global__ __launch_bounds__(256) void paba_attn_kernel(
    const float2* __restrict__ r, const float* __restrict__ thr_arr,
    float* __restrict__ out, int S, int tps /* tiles per side */,
    long long totalTiles) {
  const int lane = threadIdx.x & 31;
  const int wave = threadIdx.x >> 5;
  long long w = (long long)blockIdx.x * 8 + wave;
  if (w >= totalTiles) return;

  int tj = (int)(w % tps);
  long long t = w / tps;
  int ti = (int)(t % tps);
  int b = (int)(t / tps);
  const int i0 = ti * 16, j0 = tj * 16;

  float* outT = out + ((long long)b * S + i0) * (long long)S + j0;

  if (tj > ti) {
    // strictly above the causal diagonal: all zeros, vectorized fill
    int row = lane >> 2;
    int col = (lane & 3) * 4;
    float4 z = make_float4(0.f, 0.f, 0.f, 0.f);
    *(float4*)(outT + (long long)row * S + col) = z;
    *(float4*)(outT + (long long)(row + 8) * S + col) = z;
    return;
  }

  const float thr = thr_arr[b];
  const float2* pb = r + (long long)b * S;

  // WMMA f32 16x16x4 operands: lanes 0-15 carry K=0,1; lanes 16-31 carry the
  // K=2,3 zero padding.
  //   A row m  : { x_i[m], 1, 0, 0 }       B col n : { 1, -x_j[n], 0, 0 }
  float xi = 0.f, yi = 0.f, xj = 0.f, yj = 0.f, one = 0.f;
  if (lane < 16) {
    float2 pi = pb[i0 + lane];
    float2 pj = pb[j0 + lane];
    xi = pi.x; yi = pi.y;
    xj = pj.x; yj = pj.y;
    one = 1.0f;
  }
  v2f Ax; Ax[0] = xi;  Ax[1] = one;
  v2f Bx; Bx[0] = one; Bx[1] = -xj;
  v2f Ay; Ay[0] = yi;  Ay[1] = one;
  v2f By; By[0] = one; By[1] = -yj;
  const v8f zc = {};
  const v8f adx = __builtin_amdgcn_wmma_f32_16x16x4_f32(
      false, Ax, false, Bx, (short)0, zc, false, false);
  const v8f ady = __builtin_amdgcn_wmma_f32_16x16x4_f32(
      false, Ay, false, By, (short)0, zc, false, false);

  const int csel = lane & 15;            // C/D layout: N = lane % 16
  const int mhi = (lane >= 16) ? 8 : 0;  // C/D layout: VGPR rr -> M = rr (+8)

  if (ti == tj) {
    paba_epilogue<true>(adx, ady, thr, outT, S, csel, mhi);
  } else {
    paba_epilogue<false>(adx, ady, thr, outT, S, csel, mhi);
  }
}

// ---------------------------------------------------------------------------
extern "C" void kernel_launch(void* const* d_in, const int* in_sizes, int n_in,
                              void* d_out, int out_size, void* d_ws,
                              size_t ws_size, hipStream_t stream) {
  const float2* r = (const float2*)d_in[0];
  float* out = (float*)d_out;
  float* thr = (float*)d_ws;

  const int B = 2;
  const int S = in_sizes[0] / (B * 2);   // [B, S, 2] float32
  const int tps = S / 16;
  const long long totalTiles = (long long)B * tps * tps;
  const int blocks = (int)((totalTiles + 7) / 8);

  paba_stats_kernel<<<B, 256, 0, stream>>>(r, thr, S);
  paba_attn_kernel<<<blocks, 256, 0, stream>>>(r, thr, out, S, tps, totalTiles);
}